// MultiConstraintFM_77438260347211
// MI455X (gfx1250) — compile-verified
//
#include <hip/hip_runtime.h>
#include <cstdint>

typedef float v2f __attribute__((ext_vector_type(2)));
typedef float v8f __attribute__((ext_vector_type(8)));
typedef int   v4i __attribute__((ext_vector_type(4)));

typedef __attribute__((address_space(1))) v4i* gptr_v4i;   // global
typedef __attribute__((address_space(3))) v4i* lptr_v4i;   // LDS

#define LMBDA      100.0f
#define AUX_LMBDA  25.0f
#define REG        1e-6f
#define EPS_EV     1e-10f

#ifndef __has_builtin
#define __has_builtin(x) 0
#endif

#if defined(__HIP_DEVICE_COMPILE__) && __has_builtin(__builtin_amdgcn_global_load_async_to_lds_b128)
#define HAVE_ASYNC_LDS 1
#else
#define HAVE_ASYNC_LDS 0
#endif

#if HAVE_ASYNC_LDS
#  if __has_builtin(__builtin_amdgcn_s_wait_asynccnt)
#    define WAIT_ASYNC(n) __builtin_amdgcn_s_wait_asynccnt(n)
#  else
#    define WAIT_ASYNC(n) asm volatile("s_wait_asynccnt %0" ::"n"(n))
#  endif
#else
#  define WAIT_ASYNC(n) do {} while (0)
#endif

// 16-byte global -> LDS copy. Async path uses the CDNA5 ASYNCcnt engine
// (no VGPR staging); fallback is a synchronous reg-staged copy.
__device__ __forceinline__ void cp16_g2l(const float* g, float* l) {
#if HAVE_ASYNC_LDS
    __builtin_amdgcn_global_load_async_to_lds_b128(
        (gptr_v4i)(uintptr_t)g,
        (lptr_v4i)(unsigned int)(uintptr_t)l,
        /*offset=*/0, /*cpol=*/0);
#else
    *(float4*)l = *(const float4*)g;
#endif
}

// nan->EPS, +inf->1e6, -inf->EPS, then clamp low at EPS
__device__ __forceinline__ float safe_ev(float e) {
    if (e != e) return EPS_EV;
    e = fminf(e, 1.0e6f);
    return fmaxf(e, EPS_EV);
}

// ---------------------------------------------------------------------------
// Stage 1: P[b] = evecs[b] (128 x 4096) * feat[b] (4096 x 512), f32 WMMA.
// Workgroup = 4 waves, each wave a 16-row band, shared 64-col B tile staged
// in LDS via double-buffered async global->LDS copies (KT=32 k-steps/chunk).
// grid = (8 n-tiles of 64, 2 m-halves of 64, 16 batches), block = 128.
// ---------------------------------------------------------------------------
#define KT 32
#define NCHUNK (4096 / KT)

__global__ __launch_bounds__(128) void proj_gemm_wmma(
    const float* __restrict__ evecs,
    const float* __restrict__ feat,
    float* __restrict__ out)
{
    __shared__ float btile[2][KT * 64];   // 2 x 8 KB double buffer

    const int b    = blockIdx.z;
    const int n0   = blockIdx.x * 64;
    const int m0   = blockIdx.y * 64 + (threadIdx.x >> 5) * 16;
    const int tid  = threadIdx.x;
    const int lane = tid & 31;

    const int mrow  = lane & 15;
    const int ncol  = lane & 15;
    const int khalf = (lane >> 4) << 1;   // 0 or 2 (ISA A-frag layout)

    const float* __restrict__ Arow = evecs + (size_t)b * 128 * 4096
                                           + (size_t)(m0 + mrow) * 4096;
    const float* __restrict__ Bb   = feat  + (size_t)b * 4096 * 512 + n0;

    // Staging map: chunk = KT x 64 floats = 512 x 16B elements; 128 threads
    // copy 4 elements each. Element e: row = e>>4, 16B-column = e&15.
    v8f acc[4] = {};

    // stage chunk 0 into buffer 0
    {
        #pragma unroll
        for (int q = 0; q < 4; ++q) {
            const int e = tid + q * 128;
            const int row = e >> 4, c16 = e & 15;
            cp16_g2l(Bb + (size_t)row * 512 + c16 * 4, &btile[0][e * 4]);
        }
    }

    for (int c = 0; c < NCHUNK; ++c) {
        const int cur = c & 1;
        if (c + 1 < NCHUNK) {
            const int k1 = (c + 1) * KT;
            #pragma unroll
            for (int q = 0; q < 4; ++q) {
                const int e = tid + q * 128;
                const int row = e >> 4, c16 = e & 15;
                cp16_g2l(Bb + (size_t)(k1 + row) * 512 + c16 * 4,
                         &btile[cur ^ 1][e * 4]);
            }
            WAIT_ASYNC(4);            // chunk c's copies complete
        } else {
            WAIT_ASYNC(0);            // final chunk: drain everything
        }
        __syncthreads();              // chunk c visible to all waves

        const float* __restrict__ bt = btile[cur];
        const int k0 = c * KT;
        #pragma unroll
        for (int kk = 0; kk < KT; kk += 4) {
            v2f a;
            a.x = Arow[k0 + kk + khalf];
            a.y = Arow[k0 + kk + khalf + 1];
            #pragma unroll
            for (int t = 0; t < 4; ++t) {
                v2f bv;
                bv.x = bt[(kk + khalf) * 64 + t * 16 + ncol];
                bv.y = bt[(kk + khalf + 1) * 64 + t * 16 + ncol];
                acc[t] = __builtin_amdgcn_wmma_f32_16x16x4_f32(
                    false, a, false, bv, (short)0, acc[t], false, false);
            }
        }
        __syncthreads();              // done reading buf before it is reused
    }

    // C/D layout: VGPR v -> row (m0 + hi*8 + v), col = n-tile + (lane&15)
    const int rbase = m0 + ((lane >> 4) << 3);
    float* __restrict__ C = out + (size_t)b * 128 * 512;
    #pragma unroll
    for (int t = 0; t < 4; ++t)
        #pragma unroll
        for (int v = 0; v < 8; ++v)
            C[(size_t)(rbase + v) * 512 + n0 + t * 16 + ncol] = acc[t][v];
}

// ---------------------------------------------------------------------------
// Stage 2: Gram matrices G[i][j] = sum_d L[i][d] * R[j][d]   (K = 512)
//   which 0: AAt (L=A,R=A)   1: BAt (L=B,R=A)
//   which 2: BBt (L=B,R=B)   3: ABt (L=A,R=B)
// grid = (8, 8, 16*4), block = 32 (1 wave), one 16x16 tile per wave.
// ---------------------------------------------------------------------------
__global__ void gram_wmma(const float* __restrict__ A,
                          const float* __restrict__ B,
                          float* __restrict__ G)
{
    const int which = blockIdx.z & 3;
    const int b     = blockIdx.z >> 2;
    const float* __restrict__ L = (which == 0 || which == 3) ? A : B;
    const float* __restrict__ R = (which <= 1) ? A : B;

    const int m0   = blockIdx.y * 16;
    const int n0   = blockIdx.x * 16;
    const int lane = threadIdx.x;
    const int mrow  = lane & 15;
    const int ncol  = lane & 15;
    const int khalf = (lane >> 4) << 1;

    const float* __restrict__ Lrow = L + (size_t)b * 128 * 512
                                       + (size_t)(m0 + mrow) * 512;
    const float* __restrict__ Rrow = R + (size_t)b * 128 * 512
                                       + (size_t)(n0 + ncol) * 512;

    v8f acc = {};
    #pragma unroll 4
    for (int k = 0; k < 512; k += 4) {
        v2f a, bv;
        a.x  = Lrow[k + khalf];
        a.y  = Lrow[k + khalf + 1];
        bv.x = Rrow[k + khalf];        // B-frag element [k][n] = R[n][k]
        bv.y = Rrow[k + khalf + 1];
        acc = __builtin_amdgcn_wmma_f32_16x16x4_f32(
            false, a, false, bv, (short)0, acc, false, false);
    }

    float* __restrict__ Gb = G + ((size_t)which * 16 + b) * 128 * 128;
    const int rbase = m0 + ((lane >> 4) << 3);
    #pragma unroll
    for (int v = 0; v < 8; ++v)
        Gb[(size_t)(rbase + v) * 128 + n0 + ncol] = acc[v];
}

// ---------------------------------------------------------------------------
// Stage 3: per (dir,b,i) solve (Gram + diag(D[b,i]) + REG*I) x = rhs.
// SPD system -> pivot-free Gauss-Jordan in LDS (128 x 129 augmented).
// grid = (128 systems, 16 batches, 2 directions), block = 128 threads.
// ---------------------------------------------------------------------------
__global__ void solve_gj(const float* __restrict__ grams,
                         const float* __restrict__ evx,
                         const float* __restrict__ evy,
                         const float* __restrict__ axx,
                         const float* __restrict__ axy,
                         float* __restrict__ out)
{
    extern __shared__ float lds[];          // 128 * 129 floats (66 KB)
    __shared__ float red[128];

    const int r   = threadIdx.x;
    const int i   = blockIdx.x;
    const int b   = blockIdx.y;
    const int dir = blockIdx.z;             // 0: Cxy, 1: Cyx

    const float* __restrict__ G   = grams + ((size_t)(dir ? 2 : 0) * 16 + b) * 16384;
    const float* __restrict__ RHS = grams + ((size_t)(dir ? 3 : 1) * 16 + b) * 16384;
    const float* __restrict__ e1  = (dir ? evy : evx) + (size_t)b * 128;  // cols
    const float* __restrict__ e2  = (dir ? evx : evy) + (size_t)b * 128;  // rows
    const float* __restrict__ a1  = (dir ? axy : axx) + (size_t)b * 128;
    const float* __restrict__ a2  = (dir ? axx : axy) + (size_t)b * 128;

    // ---- per-batch scales (max over both eigenvalue sets) ----
    red[r] = fmaxf(safe_ev(e1[r]), safe_ev(e2[r]));
    __syncthreads();
    for (int s = 64; s > 0; s >>= 1) {
        if (r < s) red[r] = fmaxf(red[r], red[r + s]);
        __syncthreads();
    }
    const float scale1 = fmaxf(red[0], 1e-10f);
    __syncthreads();
    red[r] = fmaxf(safe_ev(a1[r]), safe_ev(a2[r]));
    __syncthreads();
    for (int s = 64; s > 0; s >>= 1) {
        if (r < s) red[r] = fmaxf(red[r], red[r + s]);
        __syncthreads();
    }
    const float scale2 = fmaxf(red[0], 1e-10f);

    // ---- diagonal penalty D[b,i,r] = LMBDA*mask + AUX_LMBDA*mask_aux ----
    float dterm;
    {
        float g1 = sqrtf(safe_ev(e1[r]) / scale1);
        float g2 = sqrtf(safe_ev(e2[i]) / scale1);
        float d1 = g1 * g1 + 1.0f, d2 = g2 * g2 + 1.0f;
        float mre = g2 / d2 - g1 / d1;
        float mim = 1.0f / d2 - 1.0f / d1;
        float m = mre * mre + mim * mim;
        m = (m != m) ? 0.0f : fminf(m, 1.0e6f);
        dterm = LMBDA * m;

        g1 = sqrtf(safe_ev(a1[r]) / scale2);
        g2 = sqrtf(safe_ev(a2[i]) / scale2);
        d1 = g1 * g1 + 1.0f; d2 = g2 * g2 + 1.0f;
        mre = g2 / d2 - g1 / d1;
        mim = 1.0f / d2 - 1.0f / d1;
        m = mre * mre + mim * mim;
        m = (m != m) ? 0.0f : fminf(m, 1.0e6f);
        dterm += AUX_LMBDA * m;
    }

    // ---- build augmented matrix [Gram + diag | rhs] in LDS (coalesced) ----
    for (int idx = r; idx < 128 * 128; idx += 128) {
        const int rr = idx >> 7, cc = idx & 127;
        lds[rr * 129 + cc] = G[idx];
    }
    lds[r * 129 + 128] = RHS[(size_t)i * 128 + r];
    __syncthreads();
    lds[r * 129 + r] += dterm + REG;

    // ---- Gauss-Jordan, one barrier per pivot (SPD: no pivoting needed) ----
    float* __restrict__ Mr = lds + r * 129;
    for (int c = 0; c < 128; ++c) {
        __syncthreads();
        const float* __restrict__ Mc = lds + c * 129;
        const float f = Mr[c] * (1.0f / Mc[c]);
        if (r != c) {
            for (int j = c + 1; j <= 128; ++j)
                Mr[j] -= f * Mc[j];
        }
    }
    __syncthreads();

    const float x = Mr[128] / Mr[r];
    out[(((size_t)dir * 16 + b) * 128 + i) * 128 + r] = x;
}

// ---------------------------------------------------------------------------
extern "C" void kernel_launch(void* const* d_in, const int* in_sizes, int n_in,
                              void* d_out, int out_size, void* d_ws, size_t ws_size,
                              hipStream_t stream) {
    (void)in_sizes; (void)n_in; (void)out_size; (void)ws_size;

    const float* feat_x  = (const float*)d_in[0];
    const float* feat_y  = (const float*)d_in[1];
    const float* evals_x = (const float*)d_in[2];
    const float* evals_y = (const float*)d_in[3];
    const float* evecs_x = (const float*)d_in[4];
    const float* evecs_y = (const float*)d_in[5];
    const float* aux_x   = (const float*)d_in[6];
    const float* aux_y   = (const float*)d_in[7];
    float* out = (float*)d_out;

    float* ws    = (float*)d_ws;
    float* A     = ws;                                // 16*128*512 f32
    float* Bm    = A  + (size_t)16 * 128 * 512;       // 16*128*512 f32
    float* grams = Bm + (size_t)16 * 128 * 512;       // 4 * 16*128*128 f32

    // Stage 1: projections A = evecs_x @ feat_x, B = evecs_y @ feat_y
    proj_gemm_wmma<<<dim3(8, 2, 16), 128, 0, stream>>>(evecs_x, feat_x, A);
    proj_gemm_wmma<<<dim3(8, 2, 16), 128, 0, stream>>>(evecs_y, feat_y, Bm);

    // Stage 2: AAt, BAt, BBt, ABt
    gram_wmma<<<dim3(8, 8, 64), 32, 0, stream>>>(A, Bm, grams);

    // Stage 3: 4096 SPD solves -> Cxy | Cyx
    solve_gj<<<dim3(128, 16, 2), 128, 128 * 129 * sizeof(float), stream>>>(
        grams, evals_x, evals_y, aux_x, aux_y, out);
}